// AgentCollisionReward_34651796144492
// MI455X (gfx1250) — compile-verified
//
#include <hip/hip_runtime.h>
#include <hip/hip_bf16.h>
#include <stdint.h>

// Problem-shape constants (fixed by setup_inputs):
//   N = 4096 agents, T_TOT = 20 steps, NUM_HIST = 4 -> T_FUT = 16 future steps,
//   Ne = 64 ego agents, agents-per-scene = 64.
#define T_TOT    20
#define NUM_HIST 4
#define T_FUT    16
#define MAX_N    4096

__device__ __forceinline__ float overlapf(float lx, float ly,
                                          float fr, float re, float le, float ri) {
    // LONG_T = LAT_T = 0
    float a = fmaxf(fr - lx, 0.0f);
    float b = fmaxf(re + lx, 0.0f);
    float c = fmaxf(le - ly, 0.0f);
    float d = fmaxf(ri + ly, 0.0f);
    return fminf(fminf(a, b), fminf(c, d));
}

__global__ __launch_bounds__(32 * T_FUT)
void AgentCollisionReward_kernel(const float* __restrict__ pos,          // (N, T_TOT, 2)
                                 const float* __restrict__ yaw,          // (N, T_TOT)
                                 const unsigned char* __restrict__ msk,  // (N, T_TOT) bool
                                 const float* __restrict__ box,          // (N, 4) f,r,l,rt
                                 const int* __restrict__ batch,          // (N)
                                 const int* __restrict__ ego_index,      // (Ne)
                                 float* __restrict__ out,                // Ne*T_FUT done.T + Ne reward
                                 int N, int Ne) {
    __shared__ float4 sbox[MAX_N];     // 64 KB, reused by all 16 waves
    __shared__ int    sbatch[MAX_N];   // 16 KB, fast same-scene rejection
    __shared__ int    sdone[T_FUT];

    const int tid = threadIdx.x;
    const int e   = blockIdx.x;

    // ---- Stage box (B128/lane) and batch (B32/lane) via CDNA5 async
    // global->LDS DMA (ASYNCcnt-tracked), fenced with s_wait_asynccnt + barrier.
    {
        const unsigned long long box_sa = (unsigned long long)(uintptr_t)box;
        const unsigned long long bat_sa = (unsigned long long)(uintptr_t)batch;
        for (int i = tid; i < N; i += blockDim.x) {
            unsigned lds_box = (unsigned)(uintptr_t)(&sbox[i]);
            unsigned off_box = (unsigned)(i * 16);
            asm volatile("global_load_async_to_lds_b128 %0, %1, %2"
                         :: "v"(lds_box), "v"(off_box), "s"(box_sa) : "memory");
            unsigned lds_bat = (unsigned)(uintptr_t)(&sbatch[i]);
            unsigned off_bat = (unsigned)(i * 4);
            asm volatile("global_load_async_to_lds_b32 %0, %1, %2"
                         :: "v"(lds_bat), "v"(off_bat), "s"(bat_sa) : "memory");
        }
    }
#if __has_builtin(__builtin_amdgcn_s_wait_asynccnt)
    __builtin_amdgcn_s_wait_asynccnt(0);
#else
    asm volatile("s_wait_asynccnt 0x0" ::: "memory");
#endif
    __syncthreads();

    const int w    = tid >> 5;          // wave id == future timestep t
    const int lane = tid & 31;
    const int t4   = NUM_HIST + w;      // index into T_TOT axis

    const int ego = ego_index[e];
    const int eb  = sbatch[ego];

    bool flag = false;                  // exists valid edge with loss > 0
    const bool emv = (msk[ego * T_TOT + t4] != 0);
    if (emv) {
        const float2* posv = (const float2*)pos;
        const float2 ep = posv[ego * T_TOT + t4];
        const float eyaw = yaw[ego * T_TOT + t4];
        float se, ce;
        __sincosf(eyaw, &se, &ce);
        const float4 ebx = sbox[ego];   // front, rear, left, right
        const float elx[4] = { ebx.x, ebx.x, -ebx.y, -ebx.y };
        const float ely[4] = { ebx.z, -ebx.w, -ebx.w, ebx.z };
        float ecx[4], ecy[4];
#pragma unroll
        for (int c = 0; c < 4; ++c) {
            ecx[c] = ep.x + ce * elx[c] - se * ely[c];
            ecy[c] = ep.y + se * elx[c] + ce * ely[c];
        }

        for (int n = lane; n < N; n += 32) {
            if (sbatch[n] != eb || n == ego) continue;    // LDS-only rejection
            if (!msk[n * T_TOT + t4]) continue;

            const float2 pn = posv[n * T_TOT + t4];
            const float yn = yaw[n * T_TOT + t4];
            float sa, ca;
            __sincosf(yn, &sa, &ca);
            const float4 bn = sbox[n];

            // A: ego corners in agent n's frame vs agent box
            float A = -3.0e38f;
#pragma unroll
            for (int c = 0; c < 4; ++c) {
                float dx = ecx[c] - pn.x, dy = ecy[c] - pn.y;
                float alx =  ca * dx + sa * dy;
                float aly = -sa * dx + ca * dy;
                A = fmaxf(A, overlapf(alx, aly, bn.x, bn.y, bn.z, bn.w));
            }
            // B: agent n's corners in ego frame vs ego box
            const float clx[4] = { bn.x, bn.x, -bn.y, -bn.y };
            const float cly[4] = { bn.z, -bn.w, -bn.w, bn.z };
            float Bv = -3.0e38f;
#pragma unroll
            for (int c = 0; c < 4; ++c) {
                float gx = pn.x + ca * clx[c] - sa * cly[c];
                float gy = pn.y + sa * clx[c] + ca * cly[c];
                float dx = gx - ep.x, dy = gy - ep.y;
                float blx =  ce * dx + se * dy;
                float bly = -se * dx + ce * dy;
                Bv = fmaxf(Bv, overlapf(blx, bly, ebx.x, ebx.y, ebx.z, ebx.w));
            }
            flag |= (fmaxf(A, Bv) > 0.0f);
        }
    }

    // Wave32 OR-reduction
    unsigned f = flag ? 1u : 0u;
#pragma unroll
    for (int s = 16; s > 0; s >>= 1)
        f |= (unsigned)__shfl_xor((int)f, s, 32);

    if (lane == 0) {
        sdone[w] = (int)f;
        out[e * T_FUT + w] = f ? 1.0f : 0.0f;     // done.T[e, t]
    }
    __syncthreads();
    if (tid == 0) {
        int any = 0;
#pragma unroll
        for (int t = 0; t < T_FUT; ++t) any |= sdone[t];
        out[Ne * T_FUT + e] = any ? 0.0f : 1.0f;  // reward[e]
    }
}

extern "C" void kernel_launch(void* const* d_in, const int* in_sizes, int n_in,
                              void* d_out, int out_size, void* d_ws, size_t ws_size,
                              hipStream_t stream) {
    (void)n_in; (void)out_size; (void)d_ws; (void)ws_size;
    const float*         pos   = (const float*)d_in[0];
    const float*         yaw   = (const float*)d_in[1];
    const unsigned char* msk   = (const unsigned char*)d_in[2];   // jnp bool -> byte
    const float*         box   = (const float*)d_in[3];
    const int*           batch = (const int*)d_in[4];
    const int*           egoi  = (const int*)d_in[5];

    const int N  = in_sizes[3] / 4;   // box is (N, 4)
    const int Ne = in_sizes[5];       // ego_index length

    AgentCollisionReward_kernel<<<Ne, 32 * T_FUT, 0, stream>>>(
        pos, yaw, msk, box, batch, egoi, (float*)d_out, N, Ne);
}